// AttnDecoderRNN_69183333204225
// MI455X (gfx1250) — compile-verified
//
#include <hip/hip_runtime.h>
#include <hip/hip_bf16.h>
#include <math.h>

typedef __attribute__((ext_vector_type(2))) float v2f;
typedef __attribute__((ext_vector_type(4))) float v4f;
typedef __attribute__((ext_vector_type(8))) float v8f;

#define INPUT_SIZE  300
#define HIDDEN_SIZE 1024
#define OUTPUT_SIZE 50257
#define SEQ_LEN     4096

// ---------------- workspace layout (float offsets) ----------------
#define WS_SCORES   0          // 4096  attention logits
#define WS_WEIGHTS  4096       // 4096  softmax(attn)
#define WS_APPLIED  8192       // 1024  attn_weights @ encoder_outputs (atomic acc)
#define WS_XVEC     9216       // 300   relu(combine)        (pad to 512)
#define WS_GI       9728       // 3072  x @ W_ih^T + b_ih
#define WS_GH       12800      // 3072  h @ W_hh^T + b_hh
#define WS_LOGITS   15872      // 50257 out projection
#define WS_SCAL     66136      // 2     {max, log(sum)} for log_softmax

// =====================================================================
// Row-major GEMV via V_WMMA_F32_16X16X4_F32.
//   y[r] = act( sum_k W[r,k] * x[k] + bias[r] )
// x = concat(x1[0:n1], x2[0:K-n1]) staged in LDS.
// One wave computes 16 output rows.
//
// Fast path: one b128 W load + one b128 LDS x load feed TWO WMMAs.
// K-slot permutation: lanes 0-15 carry global K {kb..kb+3}, lanes 16-31
// carry {kb+4..kb+7}; WMMA1 reduces {kb,kb+1,kb+4,kb+5}, WMMA2 the rest.
// A and B use the same permutation, so the dot product is exact.
// NT=1 marks W loads non-temporal (single-use stream larger than L2).
// =====================================================================
template <int NT>
__global__ void gemv_wmma_rows(const float* __restrict__ W,
                               const float* __restrict__ x1, int n1,
                               const float* __restrict__ x2,
                               const float* __restrict__ bias,
                               float* __restrict__ y,
                               int M, int K, int relu)
{
    extern __shared__ float sx[];
    for (int i = threadIdx.x; i < K; i += blockDim.x)
        sx[i] = (i < n1) ? x1[i] : x2[i - n1];
    __syncthreads();

    const int lane  = threadIdx.x & 31;
    const int wave  = threadIdx.x >> 5;
    const int r0    = (blockIdx.x * (blockDim.x >> 5) + wave) * 16;
    if (r0 >= M) return;                       // wave-uniform; EXEC stays all-1s

    const int m     = lane & 15;               // A-matrix row within tile
    const int khalf = lane >> 4;               // 0: lanes 0-15, 1: lanes 16-31
    int row = r0 + m; if (row >= M) row = M - 1;          // clamp tail loads
    const float* Wr4 = W + (size_t)row * (size_t)K + (size_t)(khalf * 4);

    v8f acc = {};
    const int K8 = K & ~7;
    #pragma unroll 4
    for (int kb = 0; kb < K8; kb += 8) {
        v4f w;
        if constexpr (NT) w = __builtin_nontemporal_load((const v4f*)(Wr4 + kb));
        else              w = *(const v4f*)(Wr4 + kb);
        const v4f xv = *(const v4f*)(sx + kb + khalf * 4);
        v2f a0, b0, a1, b1;
        a0.x = w.x;  a0.y = w.y;   b0.x = xv.x; b0.y = xv.y;
        acc = __builtin_amdgcn_wmma_f32_16x16x4_f32(
                  false, a0, false, b0, (short)0, acc, false, false);
        a1.x = w.z;  a1.y = w.w;   b1.x = xv.z; b1.y = xv.w;
        acc = __builtin_amdgcn_wmma_f32_16x16x4_f32(
                  false, a1, false, b1, (short)0, acc, false, false);
    }
    if (K8 < K) {                               // remainder chunk of 4 (K%4==0)
        const float* Wr2 = W + (size_t)row * (size_t)K + (size_t)(khalf * 2);
        v2f a, b;
        a.x = Wr2[K8];  a.y = Wr2[K8 + 1];
        b.x = sx[K8 + khalf * 2];  b.y = sx[K8 + khalf * 2 + 1];
        acc = __builtin_amdgcn_wmma_f32_16x16x4_f32(
                  false, a, false, b, (short)0, acc, false, false);
    }

    // D layout: lane 0 (N=0) holds M=0..7 in acc[0..7]; lane 16 holds M=8..15.
    if (m == 0) {
        const int base = r0 + khalf * 8;
        #pragma unroll
        for (int i = 0; i < 8; ++i) {
            int r = base + i;
            if (r < M) {
                float v = acc[i] + bias[r];
                if (relu) v = fmaxf(v, 0.0f);
                y[r] = v;
            }
        }
    }
}

// =====================================================================
// Column GEMV: y[h] += sum_j x[j] * E[j,h]   (E row-major [J,H])
// B = 4x16 tile of E (coalesced row segments), A = x broadcast over M.
// grid.y splits J; partials combined with global_atomic_add_f32.
// =====================================================================
__global__ void gemv_wmma_cols(const float* __restrict__ E,
                               const float* __restrict__ x,
                               float* __restrict__ y, int J, int H)
{
    extern __shared__ float sx[];
    const int jlen = J / gridDim.y;
    const int j0   = blockIdx.y * jlen;
    for (int i = threadIdx.x; i < jlen; i += blockDim.x) sx[i] = x[j0 + i];
    __syncthreads();

    const int lane  = threadIdx.x & 31;
    const int wave  = threadIdx.x >> 5;
    const int h0    = (blockIdx.x * (blockDim.x >> 5) + wave) * 16;
    if (h0 >= H) return;

    const int n     = lane & 15;
    const int khalf = lane >> 4;

    v8f acc = {};
    #pragma unroll 4
    for (int jb = 0; jb < jlen; jb += 4) {
        const int k0 = jb + khalf * 2;
        v2f a, b;
        a.x = sx[k0];                                   // x broadcast over M rows
        a.y = sx[k0 + 1];
        b.x = E[(size_t)(j0 + k0)     * H + h0 + n];    // E[j, h0+n] coalesced
        b.y = E[(size_t)(j0 + k0 + 1) * H + h0 + n];
        acc = __builtin_amdgcn_wmma_f32_16x16x4_f32(
                  false, a, false, b, (short)0, acc, false, false);
    }
    // every M row of D equals the result for column h0+n; take M=0 (acc[0], lanes 0-15)
    if (lane < 16) atomicAdd(&y[h0 + n], acc[0]);
}

// ---------------- small helper kernels ----------------
__global__ void zero_f32(float* p, int n) {
    int i = blockIdx.x * blockDim.x + threadIdx.x;
    if (i < n) p[i] = 0.0f;
}

// single-block softmax over N elems; writes normalized weights to two dests
__global__ void softmax_attn(const float* __restrict__ s,
                             float* __restrict__ w_ws,
                             float* __restrict__ w_out, int N)
{
    __shared__ float red[256];
    float m = -INFINITY;
    for (int i = threadIdx.x; i < N; i += blockDim.x) m = fmaxf(m, s[i]);
    red[threadIdx.x] = m; __syncthreads();
    for (int st = blockDim.x >> 1; st > 0; st >>= 1) {
        if (threadIdx.x < st) red[threadIdx.x] = fmaxf(red[threadIdx.x], red[threadIdx.x + st]);
        __syncthreads();
    }
    const float gmax = red[0]; __syncthreads();

    float sum = 0.0f;
    for (int i = threadIdx.x; i < N; i += blockDim.x) sum += __expf(s[i] - gmax);
    red[threadIdx.x] = sum; __syncthreads();
    for (int st = blockDim.x >> 1; st > 0; st >>= 1) {
        if (threadIdx.x < st) red[threadIdx.x] += red[threadIdx.x + st];
        __syncthreads();
    }
    const float inv = 1.0f / red[0];
    for (int i = threadIdx.x; i < N; i += blockDim.x) {
        float v = __expf(s[i] - gmax) * inv;
        w_ws[i]  = v;
        w_out[i] = v;
    }
}

// PyTorch GRU cell gate math (single step, batch 1)
__global__ void gru_gate(const float* __restrict__ gi, const float* __restrict__ gh,
                         const float* __restrict__ h, float* __restrict__ h_new, int H)
{
    int j = blockIdx.x * blockDim.x + threadIdx.x;
    if (j >= H) return;
    const float ir = gi[j],         hr = gh[j];
    const float iz = gi[H + j],     hz = gh[H + j];
    const float in_ = gi[2*H + j],  hn = gh[2*H + j];
    const float r = 1.0f / (1.0f + __expf(-(ir + hr)));
    const float z = 1.0f / (1.0f + __expf(-(iz + hz)));
    const float n = tanhf(in_ + r * hn);
    h_new[j] = (1.0f - z) * n + z * h[j];
}

// single-block reduce: scal[0]=max(logits), scal[1]=log(sum exp(logits-max))
__global__ void logsoftmax_reduce(const float* __restrict__ logits, int N,
                                  float* __restrict__ scal)
{
    __shared__ float red[256];
    float m = -INFINITY;
    for (int i = threadIdx.x; i < N; i += blockDim.x) m = fmaxf(m, logits[i]);
    red[threadIdx.x] = m; __syncthreads();
    for (int st = blockDim.x >> 1; st > 0; st >>= 1) {
        if (threadIdx.x < st) red[threadIdx.x] = fmaxf(red[threadIdx.x], red[threadIdx.x + st]);
        __syncthreads();
    }
    const float gmax = red[0]; __syncthreads();

    float sum = 0.0f;
    for (int i = threadIdx.x; i < N; i += blockDim.x) sum += __expf(logits[i] - gmax);
    red[threadIdx.x] = sum; __syncthreads();
    for (int st = blockDim.x >> 1; st > 0; st >>= 1) {
        if (threadIdx.x < st) red[threadIdx.x] += red[threadIdx.x + st];
        __syncthreads();
    }
    if (threadIdx.x == 0) { scal[0] = gmax; scal[1] = __logf(red[0]); }
}

__global__ void logsoftmax_write(const float* __restrict__ logits,
                                 const float* __restrict__ scal,
                                 float* __restrict__ out, int N)
{
    int i = blockIdx.x * blockDim.x + threadIdx.x;
    if (i < N) out[i] = logits[i] - scal[0] - scal[1];
}

// =====================================================================
extern "C" void kernel_launch(void* const* d_in, const int* in_sizes, int n_in,
                              void* d_out, int out_size, void* d_ws, size_t ws_size,
                              hipStream_t stream) {
    const float* embedded = (const float*)d_in[0];   // [300]
    const float* hidden   = (const float*)d_in[1];   // [1024]
    const float* enc      = (const float*)d_in[2];   // [4096,1024]
    const float* attn_W   = (const float*)d_in[3];   // [4096,1324]
    const float* attn_b   = (const float*)d_in[4];   // [4096]
    const float* comb_W   = (const float*)d_in[5];   // [300,1324]
    const float* comb_b   = (const float*)d_in[6];   // [300]
    const float* W_ih     = (const float*)d_in[7];   // [3072,300]
    const float* W_hh     = (const float*)d_in[8];   // [3072,1024]
    const float* b_ih     = (const float*)d_in[9];   // [3072]
    const float* b_hh     = (const float*)d_in[10];  // [3072]
    const float* out_W    = (const float*)d_in[11];  // [50257,1024]
    const float* out_b    = (const float*)d_in[12];  // [50257]

    float* out = (float*)d_out;                 // [50257 | 1024 | 4096]
    float* ws  = (float*)d_ws;

    float* ws_scores  = ws + WS_SCORES;
    float* ws_weights = ws + WS_WEIGHTS;
    float* ws_applied = ws + WS_APPLIED;
    float* ws_x       = ws + WS_XVEC;
    float* ws_gi      = ws + WS_GI;
    float* ws_gh      = ws + WS_GH;
    float* ws_logits  = ws + WS_LOGITS;
    float* ws_scal    = ws + WS_SCAL;

    float* out_logsm  = out;                        // [50257]
    float* out_hnew   = out + OUTPUT_SIZE;          // [1024]
    float* out_attnw  = out + OUTPUT_SIZE + HIDDEN_SIZE;  // [4096]

    const int IH = INPUT_SIZE + HIDDEN_SIZE;        // 1324
    const int ROWS_PER_BLK = (256 / 32) * 16;       // 8 waves * 16 rows = 128

    // 0) zero the atomic accumulator (graph replays must be deterministic)
    zero_f32<<<(HIDDEN_SIZE + 255) / 256, 256, 0, stream>>>(ws_applied, HIDDEN_SIZE);

    // 1) attention scores: [4096] = attn_W @ cat(embedded, hidden) + attn_b
    gemv_wmma_rows<0><<<(SEQ_LEN + ROWS_PER_BLK - 1) / ROWS_PER_BLK, 256,
                        IH * sizeof(float), stream>>>(
        attn_W, embedded, INPUT_SIZE, hidden, attn_b, ws_scores, SEQ_LEN, IH, 0);

    // 2) softmax -> attn_weights (ws + output slot 3)
    softmax_attn<<<1, 256, 0, stream>>>(ws_scores, ws_weights, out_attnw, SEQ_LEN);

    // 3) attn_applied[1024] = attn_weights @ encoder_outputs   (j split in 4)
    {
        dim3 grid(HIDDEN_SIZE / ROWS_PER_BLK, 4);   // (8, 4)
        gemv_wmma_cols<<<grid, 256, (SEQ_LEN / 4) * sizeof(float), stream>>>(
            enc, ws_weights, ws_applied, SEQ_LEN, HIDDEN_SIZE);
    }

    // 4) x = relu(combine_W @ cat(embedded, attn_applied) + combine_b)
    gemv_wmma_rows<0><<<(INPUT_SIZE + ROWS_PER_BLK - 1) / ROWS_PER_BLK, 256,
                        IH * sizeof(float), stream>>>(
        comb_W, embedded, INPUT_SIZE, ws_applied, comb_b, ws_x, INPUT_SIZE, IH, 1);

    // 5) GRU matmuls: gi = W_ih @ x + b_ih ; gh = W_hh @ h + b_hh
    gemv_wmma_rows<0><<<(3 * HIDDEN_SIZE + ROWS_PER_BLK - 1) / ROWS_PER_BLK, 256,
                        INPUT_SIZE * sizeof(float), stream>>>(
        W_ih, ws_x, INPUT_SIZE, nullptr, b_ih, ws_gi, 3 * HIDDEN_SIZE, INPUT_SIZE, 0);
    gemv_wmma_rows<0><<<(3 * HIDDEN_SIZE + ROWS_PER_BLK - 1) / ROWS_PER_BLK, 256,
                        HIDDEN_SIZE * sizeof(float), stream>>>(
        W_hh, hidden, HIDDEN_SIZE, nullptr, b_hh, ws_gh, 3 * HIDDEN_SIZE, HIDDEN_SIZE, 0);

    // 6) GRU gates -> h_new (output slot 2)
    gru_gate<<<(HIDDEN_SIZE + 255) / 256, 256, 0, stream>>>(
        ws_gi, ws_gh, hidden, out_hnew, HIDDEN_SIZE);

    // 7) logits = out_W @ h_new + out_b   (dominant 206 MB stream, non-temporal)
    gemv_wmma_rows<1><<<(OUTPUT_SIZE + ROWS_PER_BLK - 1) / ROWS_PER_BLK, 256,
                        HIDDEN_SIZE * sizeof(float), stream>>>(
        out_W, out_hnew, HIDDEN_SIZE, nullptr, out_b, ws_logits, OUTPUT_SIZE, HIDDEN_SIZE, 0);

    // 8) log_softmax -> output slot 1
    logsoftmax_reduce<<<1, 256, 0, stream>>>(ws_logits, OUTPUT_SIZE, ws_scal);
    logsoftmax_write<<<(OUTPUT_SIZE + 255) / 256, 256, 0, stream>>>(
        ws_logits, ws_scal, out_logsm, OUTPUT_SIZE);
}